// HybridSNNANNDetector_79645873537487
// MI455X (gfx1250) — compile-verified
//
#include <hip/hip_runtime.h>
#include <cstdint>
#include <cstddef>

#define BATCH 64
#define NA 33600          // 160*160 + 80*80 + 40*40
#define N0 25600
#define N1 6400
#define TOPK_N 500
#define KP 512            // padded top-k (16-tile aligned)
#define SCORE_THRESH 0.3f
#define NMS_THRESH 0.5f
#define CLASS_OFFSET 10000.0f

typedef __attribute__((ext_vector_type(2))) float    v2f;
typedef __attribute__((ext_vector_type(8))) float    v8f;
typedef __attribute__((ext_vector_type(4))) unsigned tdm_g0_t;
typedef __attribute__((ext_vector_type(8))) unsigned tdm_g1_t;

__device__ __forceinline__ float sigf(float x) { return 1.0f / (1.0f + expf(-x)); }

// -------------------------------------------------------------------------
// Kernel 1: per-anchor masked max score.  sc[b*NA + a] = max_c score or -1.
// -------------------------------------------------------------------------
__global__ void decode_scores(const float* __restrict__ c0, const float* __restrict__ o0,
                              const float* __restrict__ c1, const float* __restrict__ o1,
                              const float* __restrict__ c2, const float* __restrict__ o2,
                              float* __restrict__ sc)
{
    int tid = blockIdx.x * blockDim.x + threadIdx.x;
    if (tid >= BATCH * NA) return;
    int b = tid / NA, a = tid - b * NA;
    int hw, W;
    const float* cls; const float* obj;
    if (a < N0)            { hw = a;            W = 160; cls = c0; obj = o0; }
    else if (a < N0 + N1)  { hw = a - N0;       W = 80;  cls = c1; obj = o1; }
    else                   { hw = a - (N0+N1);  W = 40;  cls = c2; obj = o2; }
    int HW = W * W;
    float so = sigf(obj[(size_t)b * HW + hw]);
    float s0 = so * sigf(cls[((size_t)b * 2 + 0) * HW + hw]);
    float s1 = so * sigf(cls[((size_t)b * 2 + 1) * HW + hw]);
    float ms = fmaxf(s0, s1);
    sc[tid] = (ms > SCORE_THRESH) ? ms : -1.0f;
}

// -------------------------------------------------------------------------
// Kernel 2: per-batch top-500 by iterative argmax; whole score array resident
// in LDS (134 KB << 320 KB WGP LDS).  Exact jax top_k tie order.
// -------------------------------------------------------------------------
__global__ void __launch_bounds__(1024)
topk_select(const float* __restrict__ sc, unsigned* __restrict__ topidx)
{
    extern __shared__ unsigned lds[];
    unsigned* skey = lds;             // NA sortable keys
    unsigned* pval = skey + NA;       // 1024 per-thread stripe maxima
    unsigned* pidx = pval + 1024;
    unsigned* rval = pidx + 1024;     // reduction scratch
    unsigned* ridx = rval + 1024;

    int b = blockIdx.x, t = threadIdx.x;
    const float* s = sc + (size_t)b * NA;

    for (int i = t; i < NA; i += 1024) {
        unsigned u = __float_as_uint(s[i]);
        u = (u & 0x80000000u) ? ~u : (u | 0x80000000u);   // order-preserving
        skey[i] = u;
    }
    __syncthreads();

    unsigned bv = 0, bi = 0xFFFFFFFFu;
    for (int i = t; i < NA; i += 1024) {
        unsigned v = skey[i];
        if (v > bv) { bv = v; bi = (unsigned)i; }         // first max = smallest idx
    }
    pval[t] = bv; pidx[t] = bi;
    __syncthreads();

    for (int it = 0; it < TOPK_N; ++it) {
        rval[t] = pval[t]; ridx[t] = pidx[t];
        __syncthreads();
        for (int stp = 512; stp > 0; stp >>= 1) {
            if (t < stp) {
                unsigned v2 = rval[t + stp], i2 = ridx[t + stp];
                if (v2 > rval[t] || (v2 == rval[t] && i2 < ridx[t])) {
                    rval[t] = v2; ridx[t] = i2;
                }
            }
            __syncthreads();
        }
        unsigned best = ridx[0];
        if (t == 0) topidx[(size_t)b * TOPK_N + it] = best;
        if (t == (int)(best & 1023u)) {                   // owner thread only
            skey[best] = 0u;
            unsigned nv = 0, ni = 0xFFFFFFFFu;
            for (int i = t; i < NA; i += 1024) {
                unsigned v = skey[i];
                if (v > nv) { nv = v; ni = (unsigned)i; }
            }
            pval[t] = nv; pidx[t] = ni;
        }
        __syncthreads();
    }
}

// -------------------------------------------------------------------------
// Kernel 3: gather top-500, recompute boxes/labels only for survivors.
// -------------------------------------------------------------------------
__global__ void gather_top(const float* __restrict__ c0, const float* __restrict__ r0, const float* __restrict__ o0,
                           const float* __restrict__ c1, const float* __restrict__ r1, const float* __restrict__ o1,
                           const float* __restrict__ c2, const float* __restrict__ r2, const float* __restrict__ o2,
                           const float* __restrict__ sc, const unsigned* __restrict__ topidx,
                           float* __restrict__ topbox, float* __restrict__ boxoff,
                           float* __restrict__ toplab, float* __restrict__ topsc,
                           int* __restrict__ validf)
{
    int tid = blockIdx.x * blockDim.x + threadIdx.x;
    if (tid >= BATCH * KP) return;
    int b = tid / KP, slot = tid - b * KP;

    float bx0 = 0.f, bx1 = 0.f, bx2 = 0.f, bx3 = 0.f, lab = 0.f, msc = 0.f;
    int val = 0;
    if (slot < TOPK_N) {
        unsigned a = topidx[(size_t)b * TOPK_N + slot];
        msc = sc[(size_t)b * NA + a];
        val = (msc > SCORE_THRESH) ? 1 : 0;
        int hw, W; float stride;
        const float* cls; const float* reg; const float* obj;
        if ((int)a < N0)           { hw = a;           W = 160; stride = 8.f;  cls = c0; reg = r0; obj = o0; }
        else if ((int)a < N0 + N1) { hw = a - N0;      W = 80;  stride = 16.f; cls = c1; reg = r1; obj = o1; }
        else                       { hw = a - (N0+N1); W = 40;  stride = 32.f; cls = c2; reg = r2; obj = o2; }
        int HW = W * W;
        float so = sigf(obj[(size_t)b * HW + hw]);
        float s0 = so * sigf(cls[((size_t)b * 2 + 0) * HW + hw]);
        float s1 = so * sigf(cls[((size_t)b * 2 + 1) * HW + hw]);
        lab = (s1 > s0) ? 1.0f : 0.0f;
        int y = hw / W, x = hw - y * W;
        float dx = reg[((size_t)b * 4 + 0) * HW + hw];
        float dy = reg[((size_t)b * 4 + 1) * HW + hw];
        float dw = reg[((size_t)b * 4 + 2) * HW + hw];
        float dh = reg[((size_t)b * 4 + 3) * HW + hw];
        float cx = ((float)x + sigf(dx)) * stride;
        float cy = ((float)y + sigf(dy)) * stride;
        float w  = expf(dw) * stride;
        float h  = expf(dh) * stride;
        bx0 = cx - 0.5f * w; bx1 = cy - 0.5f * h; bx2 = cx + 0.5f * w; bx3 = cy + 0.5f * h;
    }
    float off = lab * CLASS_OFFSET;
    size_t p = (size_t)b * KP + slot;
    topbox[p*4+0] = bx0; topbox[p*4+1] = bx1; topbox[p*4+2] = bx2; topbox[p*4+3] = bx3;
    boxoff[p*4+0] = bx0 + off; boxoff[p*4+1] = bx1 + off;
    boxoff[p*4+2] = bx2 + off; boxoff[p*4+3] = bx3 + off;
    toplab[p] = lab;
    topsc[p]  = (slot < TOPK_N) ? msc : 0.f;
    validf[p] = val;
}

// -------------------------------------------------------------------------
// Kernel 4: 16x16 IoU tiles.  area_i + area_j computed as a rank-2 outer
// product on the matrix pipe (v_wmma_f32_16x16x4_f32, exact f32);
// intersection/divide finished on the VALU using the matching 16x16 f32
// accumulator layout.  One wave per tile; EXEC is all-ones at the WMMA.
// -------------------------------------------------------------------------
__global__ void __launch_bounds__(32)
iou_tiles(const float* __restrict__ boxoff, float* __restrict__ iou)
{
    int lane = threadIdx.x;
    int blk  = blockIdx.x;           // b*1024 + ti*32 + tj
    int b    = blk >> 10;
    int ti   = (blk >> 5) & 31;
    int tj   = blk & 31;
    const float* bb = boxoff + (size_t)b * KP * 4;

    __shared__ float rbox[16][4], cbox[16][4];
    __shared__ float rarea[16], carea[16];
    int l15 = lane & 15;
    if (lane < 16) {
        int ri = ti * 16 + lane;
        float x1 = bb[ri*4+0], y1 = bb[ri*4+1], x2 = bb[ri*4+2], y2 = bb[ri*4+3];
        rbox[lane][0] = x1; rbox[lane][1] = y1; rbox[lane][2] = x2; rbox[lane][3] = y2;
        rarea[lane] = (x2 - x1) * (y2 - y1);
    } else {
        int ci = tj * 16 + l15;
        float x1 = bb[ci*4+0], y1 = bb[ci*4+1], x2 = bb[ci*4+2], y2 = bb[ci*4+3];
        cbox[l15][0] = x1; cbox[l15][1] = y1; cbox[l15][2] = x2; cbox[l15][3] = y2;
        carea[l15] = (x2 - x1) * (y2 - y1);
    }
    __syncthreads();

    // A row M = [area_i, 1, 0, 0]; B col N = [1, area_j, 0, 0] over K=4.
    // f32 A 16x4 layout: lanes 0-15 hold K=0,1 in the 2 VGPRs; lanes 16-31
    // hold K=2,3 (zero here).  B mirrored.
    v2f av = {0.f, 0.f};
    v2f bvv = {0.f, 0.f};
    if (lane < 16) {
        av.x  = rarea[lane];  av.y  = 1.0f;
        bvv.x = 1.0f;         bvv.y = carea[lane];
    }
    v8f acc = {};
    acc = __builtin_amdgcn_wmma_f32_16x16x4_f32(
        /*neg_a=*/false, av, /*neg_b=*/false, bvv,
        /*c_mod=*/(short)0, acc, /*reuse_a=*/false, /*reuse_b=*/false);

    // C/D layout: lane = N (mod 16), VGPR r -> M = r (+8 for lanes 16-31).
    int N  = l15;
    int mb = (lane < 16) ? 0 : 8;
    float cx1 = cbox[N][0], cy1 = cbox[N][1], cx2 = cbox[N][2], cy2 = cbox[N][3];
    float* orow = iou + (size_t)b * KP * KP;
#pragma unroll
    for (int r = 0; r < 8; ++r) {
        int M = mb + r;
        float x1 = fmaxf(rbox[M][0], cx1), y1 = fmaxf(rbox[M][1], cy1);
        float x2 = fminf(rbox[M][2], cx2), y2 = fminf(rbox[M][3], cy2);
        float inter = fmaxf(x2 - x1, 0.f) * fmaxf(y2 - y1, 0.f);
        float uni   = acc[r] - inter;     // area_i + area_j - inter
        orow[(size_t)(ti * 16 + M) * KP + (tj * 16 + N)] = inter / (uni + 1e-7f);
    }
}

// -------------------------------------------------------------------------
// Kernel 5: greedy NMS scan + final outputs.  One block per batch.
// Box tile (8 KB, contiguous) staged into LDS by the Tensor Data Mover
// (tensor_load_to_lds, TENSORcnt); sequential scan hides row latency with
// global_prefetch of row i+1.
// -------------------------------------------------------------------------
__global__ void __launch_bounds__(KP)
nms_scan(const float* __restrict__ topbox, const float* __restrict__ boxoff,
         const float* __restrict__ toplab, const float* __restrict__ topsc,
         const int* __restrict__ validf, const float* __restrict__ iou,
         int useIou, float* __restrict__ out)
{
    __shared__ float sbox[KP][4];
    __shared__ int   svalid[KP];
    __shared__ int   ssupp[KP];
    __shared__ int   skeeparr[KP];
    __shared__ int   skeep;

    int b = blockIdx.x, j = threadIdx.x;
    size_t base = (size_t)b * KP;

    // --- TDM: DMA this batch's 512x4 f32 boxoff tile into LDS (wave 0 only;
    // TDM ignores EXEC, one descriptor per workgroup). D#: count=1, type=2,
    // data_size=4B (code 2), tile 2048x1 elements, contiguous.
    if (j < 32) {
        unsigned long long ga = (unsigned long long)(const void*)(boxoff + base * 4);
        unsigned ldsa = (unsigned)(unsigned long long)(void*)&sbox[0][0];
        tdm_g0_t g0;
        g0[0] = 1u;                                        // count=1 (valid user D#)
        g0[1] = ldsa;                                      // lds_addr
        g0[2] = (unsigned)(ga & 0xFFFFFFFFu);              // global_addr[31:0]
        g0[3] = (unsigned)((ga >> 32) & 0x01FFFFFFu)       // global_addr[56:32]
                | 0x80000000u;                             // type=2 ("image")
        tdm_g1_t g1;
        g1[0] = 0x20000u;                                  // data_size=2 (4B), mask=0
        g1[1] = (2048u << 16);                             // tensor_dim0[15:0] -> bits 63:48
        g1[2] = (1u << 16);                                // tensor_dim0 hi=0 | tensor_dim1=1
        g1[3] = (2048u << 16);                             // tensor_dim1 hi=0 | tile_dim0=2048
        g1[4] = 1u;                                        // tile_dim1=1, tile_dim2=0
        g1[5] = 2048u;                                     // tensor_dim0_stride lo
        g1[6] = 0u;
        g1[7] = 0u;
        asm volatile("tensor_load_to_lds %0, %1" : : "s"(g0), "s"(g1) : "memory");
        __builtin_amdgcn_s_wait_tensorcnt(0);
    }
    svalid[j]   = validf[base + j];
    ssupp[j]    = 0;
    skeeparr[j] = 0;
    __syncthreads();

    float ax1 = sbox[j][0], ay1 = sbox[j][1], ax2 = sbox[j][2], ay2 = sbox[j][3];
    float aarea = (ax2 - ax1) * (ay2 - ay1);
    const float* irows = iou + (size_t)b * KP * KP;

    for (int i = 0; i < TOPK_N; ++i) {
        if (j == 0) {
            int k = (svalid[i] && !ssupp[i]) ? 1 : 0;
            skeep = k; skeeparr[i] = k;
        }
        if (useIou && i + 1 < TOPK_N)
            __builtin_prefetch(irows + (size_t)(i + 1) * KP + j, 0, 0);  // hide row latency
        __syncthreads();
        if (skeep) {
            float v;
            if (useIou) {
                v = irows[(size_t)i * KP + j];
            } else {
                float bx1 = sbox[i][0], by1 = sbox[i][1], bx2 = sbox[i][2], by2 = sbox[i][3];
                float x1 = fmaxf(ax1, bx1), y1 = fmaxf(ay1, by1);
                float x2 = fminf(ax2, bx2), y2 = fminf(ay2, by2);
                float inter = fmaxf(x2 - x1, 0.f) * fmaxf(y2 - y1, 0.f);
                float barea = (bx2 - bx1) * (by2 - by1);
                v = inter / (aarea + barea - inter + 1e-7f);
            }
            if (v >= NMS_THRESH) ssupp[j] = 1;
        }
        __syncthreads();
    }

    if (j < TOPK_N) {
        int keep = skeeparr[j];
        float* ob = out;                                    // boxes  B*500*4
        float* os = out + (size_t)BATCH * TOPK_N * 4;       // scores B*500
        float* ol = os  + (size_t)BATCH * TOPK_N;           // labels B*500
        float* ok = ol  + (size_t)BATCH * TOPK_N;           // keep   B*500
        size_t p = (size_t)b * TOPK_N + j;
        const float* tb = topbox + (base + j) * 4;
        ob[p*4+0] = keep ? tb[0] : 0.f;
        ob[p*4+1] = keep ? tb[1] : 0.f;
        ob[p*4+2] = keep ? tb[2] : 0.f;
        ob[p*4+3] = keep ? tb[3] : 0.f;
        os[p] = keep ? topsc[base + j] : 0.f;
        ol[p] = toplab[base + j];
        ok[p] = keep ? 1.0f : 0.0f;
    }
}

// -------------------------------------------------------------------------
extern "C" void kernel_launch(void* const* d_in, const int* in_sizes, int n_in,
                              void* d_out, int out_size, void* d_ws, size_t ws_size,
                              hipStream_t stream)
{
    (void)in_sizes; (void)n_in; (void)out_size;
    const float* c0 = (const float*)d_in[0];
    const float* r0 = (const float*)d_in[1];
    const float* o0 = (const float*)d_in[2];
    const float* c1 = (const float*)d_in[3];
    const float* r1 = (const float*)d_in[4];
    const float* o1 = (const float*)d_in[5];
    const float* c2 = (const float*)d_in[6];
    const float* r2 = (const float*)d_in[7];
    const float* o2 = (const float*)d_in[8];
    float* out = (float*)d_out;
    uint8_t* ws = (uint8_t*)d_ws;

    size_t off = 0;
    auto take = [&](size_t bytes) -> size_t {
        size_t o = off; off = (off + bytes + 255) & ~(size_t)255; return o;
    };
    size_t o_sc   = take((size_t)BATCH * NA * 4);
    size_t o_tidx = take((size_t)BATCH * TOPK_N * 4);
    size_t o_tbox = take((size_t)BATCH * KP * 4 * 4);
    size_t o_boff = take((size_t)BATCH * KP * 4 * 4);
    size_t o_tlab = take((size_t)BATCH * KP * 4);
    size_t o_tsc  = take((size_t)BATCH * KP * 4);
    size_t o_val  = take((size_t)BATCH * KP * 4);
    size_t o_iou  = take((size_t)BATCH * KP * KP * 4);
    int useIou = (ws_size >= off) ? 1 : 0;   // deterministic: depends only on ws_size

    float*    sc   = (float*)(ws + o_sc);
    unsigned* tidx = (unsigned*)(ws + o_tidx);
    float*    tbox = (float*)(ws + o_tbox);
    float*    boff = (float*)(ws + o_boff);
    float*    tlab = (float*)(ws + o_tlab);
    float*    tsc  = (float*)(ws + o_tsc);
    int*      val  = (int*)(ws + o_val);
    float*    iou  = (float*)(ws + o_iou);

    decode_scores<<<(BATCH * NA + 255) / 256, 256, 0, stream>>>(c0, o0, c1, o1, c2, o2, sc);

    size_t topk_lds = (size_t)NA * 4 + 4 * 1024 * 4;   // 150,784 B < 320 KB WGP LDS
    topk_select<<<BATCH, 1024, topk_lds, stream>>>(sc, tidx);

    gather_top<<<(BATCH * KP + 255) / 256, 256, 0, stream>>>(
        c0, r0, o0, c1, r1, o1, c2, r2, o2, sc, tidx, tbox, boff, tlab, tsc, val);

    if (useIou)
        iou_tiles<<<BATCH * 1024, 32, 0, stream>>>(boff, iou);

    nms_scan<<<BATCH, KP, 0, stream>>>(tbox, boff, tlab, tsc, val, iou, useIou, out);
}